// CrossAttentionModule_73632919323387
// MI455X (gfx1250) — compile-verified
//
#include <hip/hip_runtime.h>
#include <hip/hip_bf16.h>
#include <float.h>
#include <type_traits>

// ---------------------------------------------------------------------------
// CDNA5 (gfx1250) wave32 WMMA f32 16x16x4 helpers
// ---------------------------------------------------------------------------
typedef float v2f __attribute__((ext_vector_type(2)));
typedef float v8f __attribute__((ext_vector_type(8)));

__device__ __forceinline__ v8f v8f_zero() {
    v8f z;
#pragma unroll
    for (int i = 0; i < 8; ++i) z[i] = 0.0f;
    return z;
}

// D = A(16x4,f32) * B(4x16,f32) + C(16x16,f32)
__device__ __forceinline__ v8f wmma_f32(v2f a, v2f b, v8f c) {
    return __builtin_amdgcn_wmma_f32_16x16x4_f32(
        /*neg_a=*/false, a, /*neg_b=*/false, b,
        /*c_mod=*/(short)0, c, /*reuse_a=*/false, /*reuse_b=*/false);
}

#define DIMD 64
#define NGRAPH 128
#define NEG_BIG (-3.0e38f)
#define SSTRIDE 18   // even -> 8B-aligned transposed reads (ds_load_b64)
#define HSTRIDE 68   // even, bank-conflict-free halves for MLP staging

// ---------------------------------------------------------------------------
// Kernel 1: per-graph row ranges of the sorted batch arrays (binary search)
// ---------------------------------------------------------------------------
__device__ __forceinline__ int lower_bound_i(const int* a, int n, int key) {
    int lo = 0, hi = n;
    while (lo < hi) {
        int mid = (lo + hi) >> 1;
        if (a[mid] < key) lo = mid + 1; else hi = mid;
    }
    return lo;
}

__global__ void ranges_kernel(const int* __restrict__ bs, const int* __restrict__ bt,
                              int n, int* __restrict__ soff, int* __restrict__ toff) {
    int t = threadIdx.x;
    if (t <= NGRAPH) soff[t] = lower_bound_i(bs, n, t);
    int u = t - 256;
    if (u >= 0 && u <= NGRAPH) toff[u] = lower_bound_i(bt, n, u);
}

// ---------------------------------------------------------------------------
// Kernel 2: block-diagonal cross attention + residual.
// One block per graph; 4 waves; each wave owns a 16-query-row tile and does
// flash-style online softmax over 16-column KV tiles using f32 WMMA.
// Column loop split into full tiles (no clamping/masking, immediate-offset
// loads) and a single masked tail tile.
// ---------------------------------------------------------------------------
__global__ __launch_bounds__(128) void attn_kernel(
    const float* __restrict__ Q, const float* __restrict__ KV,
    const int* __restrict__ qoff, const int* __restrict__ koff,
    float* __restrict__ out)
{
    __shared__ float ldsS[4][16 * SSTRIDE];   // per-wave S tile staging

    const int g  = blockIdx.x;
    const int qs = qoff[g], qe = qoff[g + 1];
    const int ks = koff[g], ke = koff[g + 1];
    const int nq = qe - qs, nk = ke - ks;
    const int tid = threadIdx.x;
    if (nq <= 0) return;

    if (nk == 0) {  // no counterpart nodes -> rows stay zero (MLP applied later)
        for (int idx = tid; idx < nq * DIMD; idx += blockDim.x)
            out[qs * DIMD + idx] = 0.0f;
        return;
    }

    const int wave = tid >> 5;
    const int lane = tid & 31;
    const int l16  = lane & 15;
    const int hi   = lane >> 4;          // 0: lanes 0-15, 1: lanes 16-31
    const int hi8  = hi * 8;
    const int ntiles = (nq + 15) >> 4;
    const int jfull  = ks + ((nk >> 4) << 4);   // end of full column tiles
    float* myS = ldsS[wave];
    // transposed-read base: lane l16 reads its row, K pairs at 4*kc + 2*hi
    const float* mySrd = myS + l16 * SSTRIDE + hi * 2;

    for (int t = wave; t < ntiles; t += 4) {
        const int m0 = qs + t * 16;

        // ---- preload Q tile in WMMA A-layout (lane l16 = row, K pairs) ----
        v2f qa[16];
        {
            int row = m0 + l16;
            bool rv = row < qe;
            const float* qp = Q + (size_t)row * DIMD + hi * 2;
#pragma unroll
            for (int kc = 0; kc < 16; ++kc) {
                qa[kc].x = rv ? qp[kc * 4 + 0] : 0.0f;
                qa[kc].y = rv ? qp[kc * 4 + 1] : 0.0f;
            }
        }

        v8f O[4];
#pragma unroll
        for (int n = 0; n < 4; ++n) O[n] = v8f_zero();
        float mrow = NEG_BIG;   // running max of row l16 (dup across halves)
        float lrow = 0.0f;      // running sum of row l16

        // ---- one 16-column KV tile; TAIL enables clamping/masking ----
        auto col_tile = [&](auto tailc, int j0) {
            constexpr bool TAIL = decltype(tailc)::value;

            // scores S = Q * K^T  (16 chained f32 WMMAs over d=64)
            v8f S = v8f_zero();
            {
                int j = j0 + l16;
                if (TAIL) j = (j < ke) ? j : (ke - 1);
                const float* kp = KV + (size_t)j * DIMD + hi * 2;
#pragma unroll
                for (int kc = 0; kc < 16; ++kc) {
                    v2f b;
                    b.x = kp[kc * 4 + 0];
                    b.y = kp[kc * 4 + 1];
                    S = wmma_f32(qa[kc], b, S);
                }
            }

            // stage S (C-layout) to LDS, read back transposed (per-lane rows)
#pragma unroll
            for (int r = 0; r < 8; ++r)
                myS[(r + hi8) * SSTRIDE + l16] = S[r];
            asm volatile("s_wait_dscnt 0" ::: "memory");

            float sv[8];
#pragma unroll
            for (int kc = 0; kc < 4; ++kc) {
                v2f tpair = *(const v2f*)(mySrd + kc * 4);
                sv[2 * kc + 0] = tpair.x;
                sv[2 * kc + 1] = tpair.y;
            }

            // (tail only) mask invalid columns; per-row max
            float mloc = NEG_BIG;
#pragma unroll
            for (int i = 0; i < 8; ++i) {
                if (TAIL) {
                    int k = (i >> 1) * 4 + hi * 2 + (i & 1);
                    sv[i] = (j0 + k < ke) ? sv[i] : NEG_BIG;
                }
                mloc = fmaxf(mloc, sv[i]);
            }
            mloc = fmaxf(mloc, __shfl_xor(mloc, 16, 32));
            float mnew  = fmaxf(mrow, mloc);
            float alpha = __expf(mrow - mnew);

            // probabilities directly in A-layout; row sum
            float pr[8];
            float ssum = 0.0f;
#pragma unroll
            for (int i = 0; i < 8; ++i) {
                float e = __expf(sv[i] - mnew);   // masked -> exact 0
                pr[i] = e;
                ssum += e;
            }
            ssum += __shfl_xor(ssum, 16, 32);
            lrow = lrow * alpha + ssum;
            mrow = mnew;

            // broadcast alpha to C-slot rows (r + 8*hi) and rescale O
#pragma unroll
            for (int r = 0; r < 8; ++r) {
                float a = __shfl(alpha, r + hi8, 32);
                O[0][r] *= a; O[1][r] *= a; O[2][r] *= a; O[3][r] *= a;
            }

            // O += P * V   (4 feature tiles x 4 k-chunks)
            v2f pa[4];
#pragma unroll
            for (int kc = 0; kc < 4; ++kc) {
                pa[kc].x = pr[2 * kc + 0];
                pa[kc].y = pr[2 * kc + 1];
            }
            if (!TAIL) {
                // one per-lane base, all offsets immediate
                const float* vb = KV + (size_t)(j0 + hi * 2) * DIMD + l16;
#pragma unroll
                for (int n = 0; n < 4; ++n) {
#pragma unroll
                    for (int kc = 0; kc < 4; ++kc) {
                        v2f b;
                        b.x = vb[(kc * 4 + 0) * DIMD + n * 16];
                        b.y = vb[(kc * 4 + 1) * DIMD + n * 16];
                        O[n] = wmma_f32(pa[kc], b, O[n]);
                    }
                }
            } else {
#pragma unroll
                for (int n = 0; n < 4; ++n) {
#pragma unroll
                    for (int kc = 0; kc < 4; ++kc) {
                        int jx = j0 + kc * 4 + hi * 2;
                        int j1 = (jx     < ke) ? jx       : (ke - 1);
                        int j2 = (jx + 1 < ke) ? (jx + 1) : (ke - 1);
                        v2f b;
                        b.x = KV[(size_t)j1 * DIMD + n * 16 + l16];
                        b.y = KV[(size_t)j2 * DIMD + n * 16 + l16];
                        O[n] = wmma_f32(pa[kc], b, O[n]);
                    }
                }
            }
        };

        for (int j0 = ks; j0 < jfull; j0 += 16)
            col_tile(std::false_type{}, j0);
        if (jfull < ke)
            col_tile(std::true_type{}, jfull);

        // ---- epilogue: normalize, add residual, store valid rows ----
        float inv = 1.0f / lrow;
#pragma unroll
        for (int r = 0; r < 8; ++r) {
            float linv = __shfl(inv, r + hi8, 32);
            int row = m0 + r + hi8;
            if (row < qe) {
#pragma unroll
                for (int n = 0; n < 4; ++n) {
                    int col = n * 16 + l16;
                    out[(size_t)row * DIMD + col] =
                        O[n][r] * linv + Q[(size_t)row * DIMD + col];
                }
            }
        }
    }
}

// ---------------------------------------------------------------------------
// Kernel 3: in-place MLP  y = relu(x@W1+b1)@W2 + b2 + x  over all rows.
// Weights staged in LDS; 4 waves per block, one 16-row tile per wave.
// ---------------------------------------------------------------------------
__global__ __launch_bounds__(128) void mlp_kernel(
    float* __restrict__ io, int nrows,
    const float* __restrict__ w1, const float* __restrict__ b1,
    const float* __restrict__ w2, const float* __restrict__ b2)
{
    __shared__ float w1s[DIMD * DIMD];
    __shared__ float w2s[DIMD * DIMD];
    __shared__ float b1s[DIMD];
    __shared__ float b2s[DIMD];
    __shared__ float hstg[4][16 * HSTRIDE];

    for (int i = threadIdx.x; i < DIMD * DIMD; i += blockDim.x) {
        w1s[i] = w1[i];
        w2s[i] = w2[i];
    }
    if (threadIdx.x < DIMD) {
        b1s[threadIdx.x] = b1[threadIdx.x];
        b2s[threadIdx.x] = b2[threadIdx.x];
    }
    __syncthreads();

    const int wave = threadIdx.x >> 5;
    const int lane = threadIdx.x & 31;
    const int l16  = lane & 15;
    const int hi   = lane >> 4;
    const int hi8  = hi * 8;
    const int ntiles = nrows >> 4;
    float* hp = hstg[wave];
    const float* hprd = hp + l16 * HSTRIDE + hi * 2;

    for (int t = blockIdx.x * 4 + wave; t < ntiles; t += gridDim.x * 4) {
        const int m0 = t * 16;

        // X tile in A-layout
        v2f xa[16];
        {
            const float* xp = io + (size_t)(m0 + l16) * DIMD + hi * 2;
#pragma unroll
            for (int kc = 0; kc < 16; ++kc) {
                xa[kc].x = xp[kc * 4 + 0];
                xa[kc].y = xp[kc * 4 + 1];
            }
        }

        // H = X @ W1
        v8f H[4];
#pragma unroll
        for (int n = 0; n < 4; ++n) H[n] = v8f_zero();
#pragma unroll
        for (int n = 0; n < 4; ++n) {
#pragma unroll
            for (int kc = 0; kc < 16; ++kc) {
                int k = kc * 4 + hi * 2;
                v2f b;
                b.x = w1s[(k + 0) * DIMD + n * 16 + l16];
                b.y = w1s[(k + 1) * DIMD + n * 16 + l16];
                H[n] = wmma_f32(xa[kc], b, H[n]);
            }
        }
        // bias + relu, stage to LDS for layout conversion
#pragma unroll
        for (int n = 0; n < 4; ++n) {
            float bb = b1s[n * 16 + l16];
#pragma unroll
            for (int r = 0; r < 8; ++r) {
                float h = H[n][r] + bb;
                h = h > 0.0f ? h : 0.0f;
                hp[(r + hi8) * HSTRIDE + n * 16 + l16] = h;
            }
        }
        asm volatile("s_wait_dscnt 0" ::: "memory");

        v2f ha[16];
#pragma unroll
        for (int kc = 0; kc < 16; ++kc)
            ha[kc] = *(const v2f*)(hprd + kc * 4);

        // Y = H @ W2
        v8f Y[4];
#pragma unroll
        for (int n = 0; n < 4; ++n) Y[n] = v8f_zero();
#pragma unroll
        for (int n = 0; n < 4; ++n) {
#pragma unroll
            for (int kc = 0; kc < 16; ++kc) {
                int k = kc * 4 + hi * 2;
                v2f b;
                b.x = w2s[(k + 0) * DIMD + n * 16 + l16];
                b.y = w2s[(k + 1) * DIMD + n * 16 + l16];
                Y[n] = wmma_f32(ha[kc], b, Y[n]);
            }
        }
        // store in place: y + b2 + x
#pragma unroll
        for (int n = 0; n < 4; ++n) {
            float bb = b2s[n * 16 + l16];
#pragma unroll
            for (int r = 0; r < 8; ++r) {
                int row = m0 + r + hi8;
                int col = n * 16 + l16;
                size_t idx = (size_t)row * DIMD + col;
                io[idx] = Y[n][r] + bb + io[idx];
            }
        }
    }
}

// ---------------------------------------------------------------------------
// Launch: ranges -> attn(tar<-src) -> attn(src<-tar) -> MLP (in-place, both)
// Output layout: d_out = [ embed_tar (N*64) | embed_src (N*64) ]
// ---------------------------------------------------------------------------
extern "C" void kernel_launch(void* const* d_in, const int* in_sizes, int n_in,
                              void* d_out, int out_size, void* d_ws, size_t ws_size,
                              hipStream_t stream) {
    (void)n_in; (void)out_size; (void)ws_size;
    const float* x_src     = (const float*)d_in[0];
    const int*   batch_src = (const int*)d_in[1];
    const float* x_tar     = (const float*)d_in[2];
    const int*   batch_tar = (const int*)d_in[3];
    const float* w1        = (const float*)d_in[4];
    const float* b1        = (const float*)d_in[5];
    const float* w2        = (const float*)d_in[6];
    const float* b2        = (const float*)d_in[7];
    const int N = in_sizes[1];

    float* out       = (float*)d_out;
    float* embed_tar = out;                       // first output of the tuple
    float* embed_src = out + (size_t)N * DIMD;    // second output

    int* soff = (int*)d_ws;        // NGRAPH+1 entries
    int* toff = soff + 132;        // NGRAPH+1 entries (aligned spacing)

    ranges_kernel<<<1, 512, 0, stream>>>(batch_src, batch_tar, N, soff, toff);
    // embed_tar = cross_attend(q=x_tar, kv=x_src, mask.T)
    attn_kernel<<<NGRAPH, 128, 0, stream>>>(x_tar, x_src, toff, soff, embed_tar);
    // embed_src = cross_attend(q=x_src, kv=x_tar, mask)
    attn_kernel<<<NGRAPH, 128, 0, stream>>>(x_src, x_tar, soff, toff, embed_src);
    // MLP applied to all rows of both tensors, in place
    mlp_kernel<<<256, 128, 0, stream>>>(out, 2 * N, w1, b1, w2, b2);
}